// batched_Matern32_37048387895518
// MI455X (gfx1250) — compile-verified
//
#include <hip/hip_runtime.h>

typedef float v2f __attribute__((ext_vector_type(2)));
typedef float v8f __attribute__((ext_vector_type(8)));

#define N_ROWS 8192
#define M_COLS 4096
#define DDIM   64
#define SQRT3F 1.7320508075688772f

// One wave computes a 32x32 output block via 4 accumulator tiles of
// V_WMMA_F32_16X16X4_F32 (pure fp32 matrix path -> matches reference fp32 math).
// Block = 256 threads = 8 waves covering 32 rows x 256 cols of the output.
__global__ __launch_bounds__(256)
void matern32_wmma_kernel(const float* __restrict__ X,
                          const float* __restrict__ Z,
                          const float* __restrict__ sigma_p,
                          const float* __restrict__ ls_p,
                          float* __restrict__ out)
{
    const int lane = threadIdx.x & 31;
    const int wave = threadIdx.x >> 5;

    const int row0 = blockIdx.y * 32;               // rows from X
    const int col0 = blockIdx.x * 256 + wave * 32;  // cols from Z

    // ---- per-lane squared row norms: lane L owns X[row0+L] and Z[col0+L] ----
    float xn = 0.f, zn = 0.f;
    {
        const float4* xr = reinterpret_cast<const float4*>(X + (size_t)(row0 + lane) * DDIM);
        const float4* zr = reinterpret_cast<const float4*>(Z + (size_t)(col0 + lane) * DDIM);
        #pragma unroll
        for (int t = 0; t < DDIM / 4; ++t) {
            float4 xv = xr[t];
            float4 zv = zr[t];
            xn += xv.x * xv.x + xv.y * xv.y + xv.z * xv.z + xv.w * xv.w;
            zn += zv.x * zv.x + zv.y * zv.y + zv.z * zv.z + zv.w * zv.w;
        }
    }

    // ---- WMMA fragment addressing (32-bit A 16x4 layout):
    // lane holds matrix-row (lane&15), K-pair base (lane>>4)*2
    const int frag_row = lane & 15;
    const int kbase    = (lane >> 4) * 2;

    const float* xa0 = X + (size_t)(row0      + frag_row) * DDIM + kbase;
    const float* xa1 = X + (size_t)(row0 + 16 + frag_row) * DDIM + kbase;
    const float* zb0 = Z + (size_t)(col0      + frag_row) * DDIM + kbase;
    const float* zb1 = Z + (size_t)(col0 + 16 + frag_row) * DDIM + kbase;

    v8f c00 = {}, c01 = {}, c10 = {}, c11 = {};

    #pragma unroll
    for (int k = 0; k < DDIM; k += 4) {
        v2f a0 = *reinterpret_cast<const v2f*>(xa0 + k);
        v2f a1 = *reinterpret_cast<const v2f*>(xa1 + k);
        v2f b0 = *reinterpret_cast<const v2f*>(zb0 + k);
        v2f b1 = *reinterpret_cast<const v2f*>(zb1 + k);
        // 8 args: (neg_a, A, neg_b, B, c_mod, C, reuse_a, reuse_b)
        c00 = __builtin_amdgcn_wmma_f32_16x16x4_f32(false, a0, false, b0, (short)0, c00, false, false);
        c01 = __builtin_amdgcn_wmma_f32_16x16x4_f32(false, a0, false, b1, (short)0, c01, false, false);
        c10 = __builtin_amdgcn_wmma_f32_16x16x4_f32(false, a1, false, b0, (short)0, c10, false, false);
        c11 = __builtin_amdgcn_wmma_f32_16x16x4_f32(false, a1, false, b1, (short)0, c11, false, false);
    }

    const float sig    = sigma_p[0];
    const float inv_ls = SQRT3F / ls_p[0];
    const float s2     = sig * sig;

    // C/D layout: VGPR r -> M = r + 8*(lane>=16), N = lane&15
    const int mhalf = (lane >> 4) * 8;
    const int ncol  = lane & 15;

    const float z2_0 = __shfl(zn, ncol,      32);  // column tile tj=0
    const float z2_1 = __shfl(zn, ncol + 16, 32);  // column tile tj=1

    #pragma unroll
    for (int r = 0; r < 8; ++r) {
        const int m0 = r + mhalf;         // row tile ti=0
        const int m1 = 16 + r + mhalf;    // row tile ti=1
        const float x2_0 = __shfl(xn, m0, 32);
        const float x2_1 = __shfl(xn, m1, 32);

        {
            float sq  = x2_0 + z2_0 - 2.f * c00[r];
            float val = inv_ls * sqrtf(fmaxf(sq, 1e-12f));
            out[(size_t)(row0 + m0) * M_COLS + (col0 + ncol)]      = s2 * (1.f + val) * __expf(-val);
        }
        {
            float sq  = x2_0 + z2_1 - 2.f * c01[r];
            float val = inv_ls * sqrtf(fmaxf(sq, 1e-12f));
            out[(size_t)(row0 + m0) * M_COLS + (col0 + 16 + ncol)] = s2 * (1.f + val) * __expf(-val);
        }
        {
            float sq  = x2_1 + z2_0 - 2.f * c10[r];
            float val = inv_ls * sqrtf(fmaxf(sq, 1e-12f));
            out[(size_t)(row0 + m1) * M_COLS + (col0 + ncol)]      = s2 * (1.f + val) * __expf(-val);
        }
        {
            float sq  = x2_1 + z2_1 - 2.f * c11[r];
            float val = inv_ls * sqrtf(fmaxf(sq, 1e-12f));
            out[(size_t)(row0 + m1) * M_COLS + (col0 + 16 + ncol)] = s2 * (1.f + val) * __expf(-val);
        }
    }
}

extern "C" void kernel_launch(void* const* d_in, const int* in_sizes, int n_in,
                              void* d_out, int out_size, void* d_ws, size_t ws_size,
                              hipStream_t stream) {
    const float* X     = (const float*)d_in[0];   // [8192, 64]
    const float* Z     = (const float*)d_in[1];   // [4096, 64]
    const float* sigma = (const float*)d_in[2];   // scalar
    const float* ls    = (const float*)d_in[3];   // scalar
    float* out = (float*)d_out;                   // [8192, 4096]

    dim3 block(256);                              // 8 waves (wave32)
    dim3 grid(M_COLS / 256, N_ROWS / 32);         // (16, 256)
    hipLaunchKernelGGL(matern32_wmma_kernel, grid, block, 0, stream,
                       X, Z, sigma, ls, out);
}